// Pen_Loss_20641612825123
// MI455X (gfx1250) — compile-verified
//
#include <hip/hip_runtime.h>
#include <cstdint>

#define AS1 __attribute__((address_space(1)))
#define AS3 __attribute__((address_space(3)))

typedef int v4i __attribute__((vector_size(16)));  // matches builtin param type

// CDNA5 async global->LDS gather path (ASYNCcnt-tracked), guarded so the
// fallback (direct b128 register gather) still compiles if the builtin is
// absent on this toolchain.
#if defined(__AMDGCN__) && __has_builtin(__builtin_amdgcn_global_load_async_to_lds_b128)
#define PEN_ASYNC 1
#else
#define PEN_ASYNC 0
#endif

#if defined(__AMDGCN__) && __has_builtin(__builtin_amdgcn_s_wait_asynccnt)
#define WAIT_ASYNC(n) __builtin_amdgcn_s_wait_asynccnt(n)
#elif defined(__AMDGCN__)
#define WAIT_ASYNC(n) asm volatile("s_wait_asynccnt %0" ::"i"(n) : "memory")
#else
#define WAIT_ASYNC(n)
#endif

namespace {
constexpr int kB = 16, kNV = 6890, kNF = 13776;
constexpr int kBP = kB / 2;          // 8 person-pair batches
constexpr int kNFP = 2 * kNF;        // 27552 triangles per pair-batch
constexpr int kNC = 65536;           // collision pairs per pair-batch
constexpr float kSigma = 1e-4f, kEps = 1e-9f, kThresh = 2000.f, kWeight = 0.1f;
constexpr int kT1 = 256;             // build kernel block
constexpr int kT2 = 256;             // pair kernel block (8 wave32)
constexpr int kIter = 4;             // pairs per lane (software pipeline depth 2)
constexpr int kBlocksPerBp = kNC / (kT2 * kIter);     // 64
constexpr int kNBlk = kBP * kBlocksPerBp;             // 512 pair blocks
constexpr size_t kTriBytes = (size_t)kBP * kNFP * 3 * sizeof(float4);
}  // namespace

struct F3 { float x, y, z; };
__device__ __forceinline__ F3 f3(const float4 a) { return {a.x, a.y, a.z}; }
__device__ __forceinline__ F3 sub3(F3 a, F3 b) { return {a.x - b.x, a.y - b.y, a.z - b.z}; }
__device__ __forceinline__ float dot3(F3 a, F3 b) { return a.x * b.x + a.y * b.y + a.z * b.z; }

// v[0..2] = intruder verts, v[3..5] = receiver verts (xyz in float4, w pad)
__device__ __forceinline__ float tri_pair_loss(const float4* v) {
  F3 i0 = f3(v[0]), i1 = f3(v[1]), i2 = f3(v[2]);
  F3 r0 = f3(v[3]), r1 = f3(v[4]), r2 = f3(v[5]);
  F3 e1 = sub3(r1, r0), e2 = sub3(r2, r0);
  F3 n = {e1.y * e2.z - e1.z * e2.y,
          e1.z * e2.x - e1.x * e2.z,
          e1.x * e2.y - e1.y * e2.x};
  float invn = 1.0f / (sqrtf(dot3(n, n)) + kEps);
  n = {n.x * invn, n.y * invn, n.z * invn};
  F3 c = {(r0.x + r1.x + r2.x) * (1.f / 3.f),
          (r0.y + r1.y + r2.y) * (1.f / 3.f),
          (r0.z + r1.z + r2.z) * (1.f / 3.f)};
  F3 q0 = sub3(r0, c), q1 = sub3(r1, c), q2 = sub3(r2, c);
  float rad = sqrtf(fmaxf(fmaxf(dot3(q0, q0), dot3(q1, q1)), dot3(q2, q2)));
  float invr = 1.0f / (rad + kEps);
  float acc = 0.f;
  F3 iv[3] = {i0, i1, i2};
#pragma unroll
  for (int t = 0; t < 3; ++t) {
    F3 dl = sub3(iv[t], c);
    float d = dot3(dl, n);
    F3 p = {dl.x - d * n.x, dl.y - d * n.y, dl.z - d * n.z};
    float radial = sqrtf(dot3(p, p));
    float f = fmaxf(-d * (1.0f / kSigma), 0.f) * fmaxf(1.f - radial * invr, 0.f);
    acc = fmaf(f, f, acc);
  }
  return acc;
}

#if PEN_ASYNC
__device__ __forceinline__ void async_b128(const float4* g, float4* l) {
  __builtin_amdgcn_global_load_async_to_lds_b128(
      (AS1 v4i*)(uintptr_t)g,                 // global src (addrspace(1))
      (AS3 v4i*)(uint32_t)(uintptr_t)l,       // LDS dst (low 32 bits = LDS offset)
      0, 0);
}
#endif

// ---- kernel 1: triangles[bp][f][v] = verts[b][faces[f%NF][v]] + trans[b] ----
__global__ __launch_bounds__(kT1) void build_tris_kernel(
    const float* __restrict__ verts, const float* __restrict__ trans,
    const int* __restrict__ faces, float4* __restrict__ tri) {
  int t = blockIdx.x * kT1 + threadIdx.x;
  if (t >= kBP * kNFP) return;
  int bp = t / kNFP;
  int f = t - bp * kNFP;
  int second = (f >= kNF) ? 1 : 0;
  int b = bp * 2 + second;
  int f0 = f - second * kNF;
  int id[3] = {faces[3 * f0 + 0], faces[3 * f0 + 1], faces[3 * f0 + 2]};
  float tx = trans[3 * b + 0], ty = trans[3 * b + 1], tz = trans[3 * b + 2];
  const float* vb = verts + (size_t)b * kNV * 3;
  float4* o = tri + (size_t)t * 3;
#pragma unroll
  for (int v = 0; v < 3; ++v) {
    const float* p = vb + 3 * (size_t)id[v];
    o[v] = make_float4(p[0] + tx, p[1] + ty, p[2] + tz, 0.f);
  }
}

// ---- kernel 2: four collision pairs per lane, pipelined async triangle gather;
//      each block writes one partial sum (deterministic, no atomics) ----
__global__ __launch_bounds__(kT2) void pen_pairs_kernel(
    const float4* __restrict__ tri, const int* __restrict__ cidx,
    float* __restrict__ partials) {
  const int tid = threadIdx.x;
  const int bp = blockIdx.x / kBlocksPerBp;
  const int cbase = (blockIdx.x - bp * kBlocksPerBp) * (kT2 * kIter);
  const float4* tb = tri + (size_t)bp * kNFP * 3;
  const int2* cb = (const int2*)(cidx + ((size_t)bp * kNC + cbase) * 2);

  float acc = 0.f;

#if PEN_ASYNC
  // Per-lane double-buffered LDS staging: 2 stages x 256 lanes x 6 float4 = 48KB.
  __shared__ float4 tbuf[2][kT2][6];
  int2 nidx = cb[tid];
  {
    const float4* gi = tb + (size_t)nidx.x * 3;
    const float4* gr = tb + (size_t)nidx.y * 3;
#pragma unroll
    for (int j = 0; j < 3; ++j) {
      async_b128(gi + j, &tbuf[0][tid][j]);
      async_b128(gr + j, &tbuf[0][tid][3 + j]);
    }
  }
#pragma unroll
  for (int k = 0; k < kIter; ++k) {
    int2 idx = nidx;
    if (k + 1 < kIter) {
      nidx = cb[(k + 1) * kT2 + tid];
      const float4* gi = tb + (size_t)nidx.x * 3;
      const float4* gr = tb + (size_t)nidx.y * 3;
      const int s = (k + 1) & 1;
#pragma unroll
      for (int j = 0; j < 3; ++j) {
        async_b128(gi + j, &tbuf[s][tid][j]);
        async_b128(gr + j, &tbuf[s][tid][3 + j]);
      }
      WAIT_ASYNC(6);  // stage k's 6 transfers done (in-order), k+1 in flight
    } else {
      WAIT_ASYNC(0);
    }
    float pl = tri_pair_loss(tbuf[k & 1][tid]);
    if (idx.x != idx.y) acc += pl;
  }
#else
#pragma unroll
  for (int k = 0; k < kIter; ++k) {
    int2 idx = cb[k * kT2 + tid];
    const float4* gi = tb + (size_t)idx.x * 3;
    const float4* gr = tb + (size_t)idx.y * 3;
    __builtin_prefetch(gi, 0, 3);
    __builtin_prefetch(gr, 0, 3);
    float4 v[6];
#pragma unroll
    for (int j = 0; j < 3; ++j) {
      v[j] = gi[j];
      v[3 + j] = gr[j];
    }
    float pl = tri_pair_loss(v);
    if (idx.x != idx.y) acc += pl;
  }
#endif

  // wave32 tree reduction (fixed tree => deterministic), then block combine
#pragma unroll
  for (int o = 16; o > 0; o >>= 1) acc += __shfl_xor(acc, o, 32);
  __shared__ float wsum[kT2 / 32];
  if ((tid & 31) == 0) wsum[tid >> 5] = acc;
  __syncthreads();
  if (tid == 0) {
    float s = 0.f;
#pragma unroll
    for (int w = 0; w < kT2 / 32; ++w) s += wsum[w];
    partials[blockIdx.x] = s;  // every slot written every launch: no init pass
  }
}

// ---- kernel 3: deterministic partial reduce + 8-element epilogue -> scalar ----
__global__ void finalize_kernel(const float* __restrict__ partials,
                                float* __restrict__ out) {
  __shared__ float pen_s[kBP];
  const int tid = threadIdx.x;
  if (tid < kBP) {
    float s = 0.f;
    for (int i = 0; i < kBlocksPerBp; ++i)      // fixed order => deterministic
      s += partials[tid * kBlocksPerBp + i];
    pen_s[tid] = s;
  }
  __syncthreads();
  if (tid == 0) {
    float cnt = 0.f, vals = 0.f;
#pragma unroll
    for (int i = 0; i < kBP; ++i) {
      float p = pen_s[i];
      float keep = (p < kThresh) ? 1.f : 0.f;
      cnt += keep;
      float s = 1.f / (1.f + __expf(-p * (1.0f / kThresh)));
      vals += (s - 0.5f) * keep;
    }
    out[0] = (cnt > 0.f) ? (vals / fmaxf(cnt, 1.f)) * kWeight : 0.f;
  }
}

extern "C" void kernel_launch(void* const* d_in, const int* in_sizes, int n_in,
                              void* d_out, int out_size, void* d_ws, size_t ws_size,
                              hipStream_t stream) {
  const float* verts = (const float*)d_in[0];   // [16, 6890, 3] f32
  const float* trans = (const float*)d_in[1];   // [16, 3] f32
  const int* faces = (const int*)d_in[2];       // [13776, 3] i32
  const int* cidx = (const int*)d_in[3];        // [8, 65536, 2] i32
  float* out = (float*)d_out;                   // scalar f32

  float4* tri = (float4*)d_ws;                              // 10.58 MB table
  float* partials = (float*)((char*)d_ws + kTriBytes);      // 512 block sums

  const int ntri = kBP * kNFP;
  hipLaunchKernelGGL(build_tris_kernel, dim3((ntri + kT1 - 1) / kT1), dim3(kT1),
                     0, stream, verts, trans, faces, tri);
  hipLaunchKernelGGL(pen_pairs_kernel, dim3(kNBlk), dim3(kT2), 0, stream, tri,
                     cidx, partials);
  hipLaunchKernelGGL(finalize_kernel, dim3(1), dim3(32), 0, stream, partials,
                     out);
}